// Embedding_TTM_order4_13322988552199
// MI455X (gfx1250) — compile-verified
//
#include <hip/hip_runtime.h>

typedef __attribute__((ext_vector_type(2))) float v2f;
typedef __attribute__((ext_vector_type(8))) float v8f;

// ---------------------------------------------------------------------------
// Table build kernels (tiny): out1[row0][b][c], out2[c][row1][d]
// ---------------------------------------------------------------------------
__global__ void build_out1(const float* __restrict__ U0,   // (1,15,8,32)
                           const float* __restrict__ U1,   // (32,15,8,64)
                           float* __restrict__ out1)       // (225,64,64)
{
    int idx = blockIdx.x * blockDim.x + threadIdx.x;
    if (idx >= 225 * 64 * 64) return;
    int c   = idx & 63;
    int b   = (idx >> 6) & 63;
    int row = idx >> 12;                 // 0..224
    int i0 = row / 15, i1 = row % 15;
    int o0 = b >> 3,   o1 = b & 7;
    const float* u0 = U0 + (i0 * 8 + o0) * 32;              // U0[0,i0,o0,k]
    const float* u1 = U1 + (i1 * 8 + o1) * 64 + c;          // U1[k,i1,o1,c]
    float s = 0.f;
#pragma unroll
    for (int k = 0; k < 32; ++k)
        s += u0[k] * u1[k * 7680];                          // 7680 = 15*8*64
    out1[idx] = s;
}

__global__ void build_out2(const float* __restrict__ U2,   // (64,15,8,32)
                           const float* __restrict__ U3,   // (32,15,8,1)
                           float* __restrict__ out2)       // (64,225,64)
{
    int idx = blockIdx.x * blockDim.x + threadIdx.x;
    if (idx >= 64 * 225 * 64) return;
    int d   = idx & 63;
    int row = (idx >> 6) % 225;
    int c   = idx / (225 * 64);
    int i2 = row / 15, i3 = row % 15;
    int o2 = d >> 3,   o3 = d & 7;
    const float* u2 = U2 + ((c * 15 + i2) * 8 + o2) * 32;   // U2[c,i2,o2,k]
    const float* u3 = U3 + (i3 * 8 + o3);                   // U3[k,i3,o3,0]
    float s = 0.f;
#pragma unroll
    for (int k = 0; k < 32; ++k)
        s += u2[k] * u3[k * 120];                           // 120 = 15*8
    out2[idx] = s;
}

// ---------------------------------------------------------------------------
// Main kernel: one block (16 waves) per sample; per-sample 64x64x64 f32 GEMM
// via V_WMMA_F32_16X16X4_F32. A/B tiles staged in LDS once per sample.
// ---------------------------------------------------------------------------
#define SA 68   // LDS stride for A (floats): bank-conflict-free strided reads

__global__ __launch_bounds__(512) void ttm_gemm_kernel(
    const int*   __restrict__ x,
    const float* __restrict__ out1,   // (225,64,64) row-major
    const float* __restrict__ out2,   // (64,225,64) row-major
    float*       __restrict__ out)    // (N,64,64)
{
    __shared__ float lA[64 * SA];
    __shared__ float lB[64 * 64];

    const int n  = blockIdx.x;
    const int t  = threadIdx.x;
    const int xi = x[n];
    const int row0 = xi / 225;
    const int row1 = xi % 225;

    // --- cooperative load: A = out1[row0] (contiguous 16KB),
    //                       B = out2[:,row1,:] (64 rows, stride 14400 floats)
    {
        const int r  = t >> 3;           // 0..63
        const int ch = (t & 7) * 8;      // 0,8,...,56
        const float4* Ag = (const float4*)(out1 + (size_t)row0 * 4096 + r * 64 + ch);
        float4 a0 = Ag[0], a1 = Ag[1];
        *(float4*)(&lA[r * SA + ch])     = a0;
        *(float4*)(&lA[r * SA + ch + 4]) = a1;
        const float4* Bg = (const float4*)(out2 + (size_t)r * 14400 + row1 * 64 + ch);
        float4 b0 = Bg[0], b1 = Bg[1];
        *(float4*)(&lB[r * 64 + ch])     = b0;
        *(float4*)(&lB[r * 64 + ch + 4]) = b1;
    }
    __syncthreads();

    // --- per-wave 16x16 tile, K=64 in 16 steps of K=4
    const int wave = t >> 5;             // 0..15
    const int lane = t & 31;
    const int tm = wave >> 2, tn = wave & 3;
    const int lr = lane & 15;            // 0..15
    const int lh = lane >> 4;            // 0 or 1

    // A frag: a[v] = A[tm*16 + lr][k0*4 + lh*2 + v]   (16x4, 2 VGPRs)
    // B frag: b[v] = B[k0*4 + lh*2 + v][tn*16 + lr]   (4x16, 2 VGPRs)
    const float* Abase = &lA[(tm * 16 + lr) * SA + lh * 2];
    const float* Bbase = &lB[(lh * 2) * 64 + tn * 16 + lr];

    v8f acc = {};
#pragma unroll
    for (int k0 = 0; k0 < 16; ++k0) {
        v2f a = *(const v2f*)(Abase + k0 * 4);
        v2f b;
        b.x = Bbase[k0 * 4 * 64];
        b.y = Bbase[k0 * 4 * 64 + 64];
        // 8 args: (neg_a, A, neg_b, B, c_mod, C, reuse_a, reuse_b)
        acc = __builtin_amdgcn_wmma_f32_16x16x4_f32(
            false, a, false, b, (short)0, acc, false, false);
    }

    // D layout: D[v + 8*lh][lr] per tile -> global row (tm*16 + lh*8 + v)
    float* op = out + (size_t)n * 4096 + (tm * 16 + lh * 8) * 64 + tn * 16 + lr;
#pragma unroll
    for (int v = 0; v < 8; ++v)
        op[v * 64] = acc[v];
}

// ---------------------------------------------------------------------------
extern "C" void kernel_launch(void* const* d_in, const int* in_sizes, int n_in,
                              void* d_out, int out_size, void* d_ws, size_t ws_size,
                              hipStream_t stream) {
    const int*   x  = (const int*)  d_in[0];
    const float* U0 = (const float*)d_in[1];
    const float* U1 = (const float*)d_in[2];
    const float* U2 = (const float*)d_in[3];
    const float* U3 = (const float*)d_in[4];
    float* out = (float*)d_out;

    float* w1 = (float*)d_ws;          // out1: 225*64*64 = 921600 floats
    float* w2 = w1 + 921600;           // out2: 64*225*64 = 921600 floats

    const int tab = 921600;
    build_out1<<<(tab + 255) / 256, 256, 0, stream>>>(U0, U1, w1);
    build_out2<<<(tab + 255) / 256, 256, 0, stream>>>(U2, U3, w2);

    const int nsamp = in_sizes[0];     // 2*4096 = 8192 samples
    ttm_gemm_kernel<<<nsamp, 512, 0, stream>>>(x, w1, w2, out);
}